// RelativePositionEncoding_87883620811381
// MI455X (gfx1250) — compile-verified
//
#include <hip/hip_runtime.h>

// ---------------------------------------------------------------------------
// RelativePositionEncoding for MI455X (gfx1250)
//
// out[i,j,:] = W_pos[d_res] + W_tok[d_tok] + same_ent*w_ent + W_chn[d_chain]
//
// Memory-bound: 512 MB f32 output @ 23.3 TB/s ~= 22 us floor. Strategy:
//   * TDM tensor_load_to_lds stages the full 139x128 f32 W table (71 KB)
//     into LDS once per workgroup (s_wait_tensorcnt to sync).
//   * Each 256-thread block (8 waves) computes an 8(i) x 128(j) pair tile;
//     wave w owns row i0+w (row scalars wave-uniform), lane covers 4 channels.
//   * 3x ds_load_b128 gathers + FMA per pair per lane, then a non-temporal
//     global_store_b128 so the 512 MB stream doesn't thrash L2.
// ---------------------------------------------------------------------------

typedef float        vf4 __attribute__((ext_vector_type(4)));
typedef unsigned int u32;
typedef u32          v4u __attribute__((ext_vector_type(4)));
typedef int          v8i __attribute__((ext_vector_type(8)));
typedef int          v4i __attribute__((ext_vector_type(4)));

#define N_TOK    1024
#define CZ       128
#define IN_DIM   139
#define W_ELEMS  (IN_DIM * CZ)        // 17792 floats
#define W_BYTES  (W_ELEMS * 4)        // 71168 bytes
#define TJ       128                  // j-tile per block
#define TI       8                    // i-tile per block (== #waves)
#define LDSJ     (5 * TJ)             // staged j-scalars (5 arrays)
#define LDS_BYTES (W_BYTES + LDSJ * 4)

__global__ __launch_bounds__(256)
void relpos_kernel(const int* __restrict__ asym, const int* __restrict__ resi,
                   const int* __restrict__ enti, const int* __restrict__ symi,
                   const int* __restrict__ toki, const float* __restrict__ W,
                   float* __restrict__ out)
{
    extern __shared__ char smem_raw[];
    float* ldsW = (float*)smem_raw;                 // [0, 71168): W table
    int*   ldsJ = (int*)(smem_raw + W_BYTES);       // 5 x 128 ints

    const int tid  = (int)threadIdx.x;
    const int wave = tid >> 5;
    const int lane = tid & 31;
    const int j0   = (int)blockIdx.x * TJ;
    const int i    = (int)blockIdx.y * TI + wave;

    if (wave == 0) {
        // --- Tensor Data Mover: DMA the whole W table to LDS offset 0 ----------
        // D# group0: count=1 (valid, user), lds_addr=0, 57-bit global addr,
        //            type=2 ("image").
        unsigned long long wa = (unsigned long long)(uintptr_t)W;
        v4u g0;
        g0[0] = 1u;
        g0[1] = 0u;                                   // lds_addr = 0 (dyn LDS base)
        g0[2] = (u32)wa;                              // global_addr[31:0]
        g0[3] = ((u32)(wa >> 32) & 0x01ffffffu)       // global_addr[56:32]
              | 0x80000000u;                          // type = 2
        // D# group1: data_size=4B; 1-D tile of W_ELEMS elements.
        v8i g1;
        g1[0] = (int)(2u << 16);                      // data_size = 2 (4 bytes)
        g1[1] = (int)((u32)W_ELEMS << 16);            // tensor_dim0[15:0]
        g1[2] = (int)(1u << 16);                      // dim0 hi=0 | tensor_dim1=1
        g1[3] = (int)((u32)W_ELEMS << 16);            // dim1 hi=0 | tile_dim0
        g1[4] = 1;                                    // tile_dim1=1, tile_dim2=0
        g1[5] = W_ELEMS;                              // tensor_dim0_stride lo32
        g1[6] = (int)((u32)W_ELEMS << 16);            // stride0 hi=0 | stride1 lo16
        g1[7] = 0;                                    // stride1 hi
        v4i g2; g2[0] = 1; g2[1] = 1; g2[2] = W_ELEMS; g2[3] = 0;   // benign dims
        v4i g3; g3[0] = W_ELEMS; g3[1] = (int)(1u << 16); g3[2] = 0; g3[3] = 0;
        v8i g4 = {0, 0, 0, 0, 0, 0, 0, 0};            // VADDR4 slot: unused/NULL
        __builtin_amdgcn_tensor_load_to_lds(g0, g1, g2, g3, g4, /*cpol=*/0);
    }

    // All 256 threads stage this block's 128 j-scalars (5 arrays) into LDS.
    // Unsigned k => a = k>>7, e = k&127 lower to plain shifts/masks.
    for (u32 k = (u32)tid; k < (u32)LDSJ; k += 256u) {
        const u32 a = k >> 7, e = k & 127u;
        const int* src = (a == 0u) ? asym : (a == 1u) ? resi
                        : (a == 2u) ? enti : (a == 3u) ? symi : toki;
        ldsJ[k] = src[j0 + (int)e];
    }

    // Wave 0 drains its TDM op; waves 1..7 pass trivially (TENSORcnt==0).
    __builtin_amdgcn_s_wait_tensorcnt(0);
    asm volatile("" ::: "memory");
    __syncthreads();

    // Row-i scalars (uniform within the wave).
    const int ai = asym[i], ri = resi[i], ei = enti[i], si = symi[i], ti = toki[i];

    const vf4* Wv   = (const vf4*)ldsW;       // row r, lane l -> Wv[r*32 + l]
    const vf4  went = Wv[132 * 32 + lane];    // w_ent row, j-invariant

    vf4*   outv = (vf4*)out;
    size_t base = ((size_t)i * N_TOK + j0) * (CZ / 4) + lane;

    #pragma unroll 4
    for (int jj = 0; jj < TJ; ++jj) {
        const int aj = ldsJ[jj];
        const int rj = ldsJ[TJ + jj];
        const int ej = ldsJ[2 * TJ + jj];
        const int sj = ldsJ[3 * TJ + jj];
        const int tj = ldsJ[4 * TJ + jj];

        const bool sc = (ai == aj);
        const bool sr = (ri == rj);
        const bool se = (ei == ej);
        const int dr = sc        ? min(max(ri - rj + 32, 0), 64) : 65;   // 0..65
        const int dt = (sc && sr)? min(max(ti - tj + 32, 0), 64) : 65;   // 0..65
        const int dc = se        ? min(max(si - sj + 2, 0), 4)  : 5;     // 0..5

        vf4 o = Wv[dr * 32 + lane];             // W_pos row
        o += Wv[(66 + dt) * 32 + lane];         // W_tok row
        o += Wv[(133 + dc) * 32 + lane];        // W_chn row
        const float fe = se ? 1.0f : 0.0f;
        o.x = fmaf(fe, went.x, o.x);
        o.y = fmaf(fe, went.y, o.y);
        o.z = fmaf(fe, went.z, o.z);
        o.w = fmaf(fe, went.w, o.w);

        __builtin_nontemporal_store(o, &outv[base + (size_t)jj * (CZ / 4)]);
    }
}

extern "C" void kernel_launch(void* const* d_in, const int* in_sizes, int n_in,
                              void* d_out, int out_size, void* d_ws, size_t ws_size,
                              hipStream_t stream) {
    const int*   asym = (const int*)d_in[0];
    const int*   resi = (const int*)d_in[1];
    const int*   enti = (const int*)d_in[2];
    const int*   symi = (const int*)d_in[3];
    const int*   toki = (const int*)d_in[4];
    const float* W    = (const float*)d_in[5];
    float*       out  = (float*)d_out;

    dim3 grid(N_TOK / TJ, N_TOK / TI);   // (8, 128)
    relpos_kernel<<<grid, 256, LDS_BYTES, stream>>>(asym, resi, enti, symi, toki, W, out);
}